// DeformableGrid_86517821214971
// MI455X (gfx1250) — compile-verified
//
#include <hip/hip_runtime.h>
#include <stdint.h>

// Problem constants (from reference): B=4, C=64, N=512*512, K=400
#define B_      4
#define C_      64
#define N_      (512 * 512)
#define K_      400
#define KP_     401            // K padded: 401 % 64 = 17 (coprime) -> conflict-free LDS banks
#define CP_     65             // C padded: row stride 65 -> bank = (k + c) % 64, spreads random k
#define CNT_OFF (C_ * KP_)     // counts region offset inside kernel1 LDS
#define CHUNK_  4096           // pixels per block
#define THREADS 256            // 8 wave32s

typedef float v4f  __attribute__((ext_vector_type(4)));
typedef int   v4i  __attribute__((ext_vector_type(4)));
typedef unsigned int u32x4 __attribute__((ext_vector_type(4)));
typedef int   i32x8 __attribute__((ext_vector_type(8)));
typedef int   i32x4 __attribute__((ext_vector_type(4)));

#if defined(__has_builtin)
#  if __has_builtin(__builtin_amdgcn_tensor_load_to_lds)
#    define USE_TDM 1
#  endif
#  if __has_builtin(__builtin_amdgcn_s_wait_tensorcnt)
#    define HAVE_WAIT_TENSOR 1
#  endif
#endif
#ifndef USE_TDM
#  define USE_TDM 0
#endif
#ifndef HAVE_WAIT_TENSOR
#  define HAVE_WAIT_TENSOR 0
#endif

// ---------------------------------------------------------------------------
// Kernel 1: per-block LDS segment accumulation (ds_add_f32), then one global
// atomic flush per block. LDS layout: sum[c][k] with row stride KP_=401 so
// bank = (17*c + k) % 64 — random k spreads across all banks; counts at end.
// ---------------------------------------------------------------------------
__global__ void __launch_bounds__(THREADS)
k_accum(const float* __restrict__ feat, const int* __restrict__ sidx,
        float* __restrict__ gsum, float* __restrict__ gcnt) {
    extern __shared__ float sm[];
    const int tid  = threadIdx.x;
    const int b    = blockIdx.y;
    const int pix0 = blockIdx.x * CHUNK_;

    for (int i = tid; i < CNT_OFF + K_; i += THREADS) sm[i] = 0.0f;
    __syncthreads();

    const int lane = tid & 31;
    const int wave = tid >> 5;
    const size_t fbase = (size_t)b * C_ * N_;
    const int* ip = sidx + (size_t)b * N_;

    // Each wave covers 128 consecutive pixels per iteration (4 per lane, b128 loads)
    for (int it = 0; it < CHUNK_; it += THREADS * 4) {
        const int n0 = pix0 + it + wave * 128 + lane * 4;
        const v4i kv = *(const v4i*)(ip + n0);
        atomicAdd(&sm[CNT_OFF + kv.x], 1.0f);
        atomicAdd(&sm[CNT_OFF + kv.y], 1.0f);
        atomicAdd(&sm[CNT_OFF + kv.z], 1.0f);
        atomicAdd(&sm[CNT_OFF + kv.w], 1.0f);
        const float* fp = feat + fbase + n0;
#pragma unroll 4
        for (int c = 0; c < C_; ++c) {
            // 256 MB stream: non-temporal so it doesn't thrash the L2-resident idx/means
            v4f v = __builtin_nontemporal_load((const v4f*)(fp + (size_t)c * N_));
            atomicAdd(&sm[c * KP_ + kv.x], v.x);
            atomicAdd(&sm[c * KP_ + kv.y], v.y);
            atomicAdd(&sm[c * KP_ + kv.z], v.z);
            atomicAdd(&sm[c * KP_ + kv.w], v.w);
        }
    }
    __syncthreads();

    // Flush partial sums: global layout [b][k][c] (contiguous, coalesced atomics
    // into a 400 KB L2-resident region)
    float* gs = gsum + (size_t)b * K_ * C_;
    for (int i = tid; i < K_ * C_; i += THREADS) {
        const int k = i / C_;
        const int c = i % C_;
        unsafeAtomicAdd(&gs[i], sm[c * KP_ + k]);
    }
    for (int i = tid; i < K_; i += THREADS)
        unsafeAtomicAdd(&gcnt[b * K_ + i], sm[CNT_OFF + i]);
}

// ---------------------------------------------------------------------------
// Kernel 2: means = sums / max(counts, 1), in place. B*K*C = 102400 elements.
// ---------------------------------------------------------------------------
__global__ void __launch_bounds__(THREADS)
k_mean(float* __restrict__ sums, const float* __restrict__ cnts) {
    const int i = blockIdx.x * THREADS + threadIdx.x;   // i = (b*K + k)*C + c
    const float c = cnts[i / C_];
    sums[i] *= (1.0f / fmaxf(c, 1.0f));
}

// ---------------------------------------------------------------------------
// Kernel 3: stage this batch's means table [K][C] into LDS with the Tensor
// Data Mover (pad_enable gives 65-float row stride for bank-free gathers),
// then per-pixel gather with coalesced non-temporal b128 stores.
// ---------------------------------------------------------------------------
__global__ void __launch_bounds__(THREADS)
k_gather(const float* __restrict__ means, const int* __restrict__ sidx,
         float* __restrict__ out) {
    extern __shared__ float sm[];
    const int tid  = threadIdx.x;
    const int b    = blockIdx.y;
    const int pix0 = blockIdx.x * CHUNK_;
    const float* mb = means + (size_t)b * K_ * C_;

#if USE_TDM
    if (tid < 32) {
        const uint64_t ga = (uint64_t)(uintptr_t)mb;
        // D# group0: count=1, lds_addr=0 (dynamic LDS base), global_addr, type=2
        u32x4 g0 = { 1u, 0u,
                     (unsigned)(ga & 0xFFFFFFFFu),
                     (unsigned)((ga >> 32) & 0x1FFFFFFu) | (2u << 30) };
        // D# group1: data_size=4B(2), pad_enable, pad_interval=5 (64 DWORDs),
        // pad_amount=0 (1 DWORD) -> LDS row stride 65 floats.
        // tensor_dim0=C=64, tensor_dim1=K=400, tile_dim0=64, tile_dim1=400,
        // tensor_dim0_stride=64.
        i32x8 g1 = { (2 << 16) | (1 << 20) | (5 << 22),
                     (C_ << 16),          // tensor_dim0 low16 @ bits[63:48]
                     (K_ << 16),          // dim0 hi16=0 | tensor_dim1 low16
                     (C_ << 16),          // dim1 hi16=0 | tile_dim0
                     K_,                  // tile_dim1 | tile_dim2=0
                     C_,                  // tensor_dim0_stride low32
                     0, 0 };
        i32x4 gz4 = { 0, 0, 0, 0 };       // groups 2/3 unused (2D tensor)
        i32x8 gz8 = { 0, 0, 0, 0, 0, 0, 0, 0 };
        __builtin_amdgcn_tensor_load_to_lds(g0, g1, gz4, gz4, gz8, 0);
#if HAVE_WAIT_TENSOR
        __builtin_amdgcn_s_wait_tensorcnt(0);
#else
        asm volatile("s_wait_tensorcnt 0" ::: "memory");
#endif
    }
#else
    // Fallback staging: coalesced global read, padded LDS write [k][65]
    for (int i = tid; i < K_ * C_; i += THREADS) {
        const int k = i / C_;
        const int c = i % C_;
        sm[k * CP_ + c] = mb[i];
    }
#endif
    __syncthreads();

    const int lane = tid & 31;
    const int wave = tid >> 5;
    const size_t obase = (size_t)b * C_ * N_;
    const int* ip = sidx + (size_t)b * N_;

    for (int it = 0; it < CHUNK_; it += THREADS * 4) {
        const int n0 = pix0 + it + wave * 128 + lane * 4;
        const v4i kv = *(const v4i*)(ip + n0);
        const int a0 = kv.x * CP_, a1 = kv.y * CP_, a2 = kv.z * CP_, a3 = kv.w * CP_;
        float* op = out + obase + n0;
#pragma unroll 4
        for (int c = 0; c < C_; ++c) {
            v4f o;
            o.x = sm[a0 + c];
            o.y = sm[a1 + c];
            o.z = sm[a2 + c];
            o.w = sm[a3 + c];
            __builtin_nontemporal_store(o, (v4f*)(op + (size_t)c * N_));
        }
    }
}

// ---------------------------------------------------------------------------
extern "C" void kernel_launch(void* const* d_in, const int* in_sizes, int n_in,
                              void* d_out, int out_size, void* d_ws, size_t ws_size,
                              hipStream_t stream) {
    (void)in_sizes; (void)n_in; (void)out_size; (void)ws_size;
    const float* feat = (const float*)d_in[0];   // [B, C, N] fp32
    const int*   sidx = (const int*)d_in[1];     // [B, N] int32
    float* out  = (float*)d_out;                 // [B, C, N] fp32
    float* sums = (float*)d_ws;                  // [B*K*C] fp32 (becomes means)
    float* cnts = sums + (size_t)B_ * K_ * C_;   // [B*K] fp32

    // Zero accumulators every call (graph-capture safe)
    (void)hipMemsetAsync(d_ws, 0, (size_t)(B_ * K_ * C_ + B_ * K_) * sizeof(float), stream);

    const dim3 grid(N_ / CHUNK_, B_);            // (64, 4)
    const size_t lds1 = (size_t)(CNT_OFF + K_) * sizeof(float);   // ~104 KB
    const size_t lds3 = (size_t)(K_ * CP_) * sizeof(float);       // ~104 KB

    k_accum<<<grid, THREADS, lds1, stream>>>(feat, sidx, sums, cnts);
    k_mean<<<(B_ * K_ * C_) / THREADS, THREADS, 0, stream>>>(sums, cnts);
    k_gather<<<grid, THREADS, lds3, stream>>>(sums, sidx, out);
}